// DPTreeOnlyKeyAttention_21079699489137
// MI455X (gfx1250) — compile-verified
//
#include <hip/hip_runtime.h>

// Problem constants (fixed by the reference harness)
#define TQ 64
#define NB 8
#define NE 512
#define NH 8
#define NT 64
#define TKN 127   // tree node count
#define ND 64     // head dim
#define NBH 64    // B*H

typedef __attribute__((ext_vector_type(2))) float v2f;
typedef __attribute__((ext_vector_type(8))) float v8f;
typedef int b128_t __attribute__((vector_size(16)));   // matches builtin param pointee

// ---------------------------------------------------------------------------
// Async global -> LDS staging (CDNA5 GLOBAL_LOAD_ASYNC_TO_LDS_B128, ASYNCcnt)
// ---------------------------------------------------------------------------
#if defined(__gfx1250__) && __has_builtin(__builtin_amdgcn_global_load_async_to_lds_b128)
#define HAS_ASYNC_LDS 1
#else
#define HAS_ASYNC_LDS 0
#endif

__device__ __forceinline__ void g2l_b128(const float* g, float* l) {
#if HAS_ASYNC_LDS
  __builtin_amdgcn_global_load_async_to_lds_b128(
      (__attribute__((address_space(1))) b128_t*)g,
      (__attribute__((address_space(3))) b128_t*)l,
      /*imm_offset=*/0, /*cpol=*/0);
#else
  *(float4*)l = *(const float4*)g;
#endif
}

__device__ __forceinline__ void async_wait0() {
#if HAS_ASYNC_LDS
#if __has_builtin(__builtin_amdgcn_s_wait_asynccnt)
  __builtin_amdgcn_s_wait_asynccnt(0);
#else
  asm volatile("s_wait_asynccnt 0" ::: "memory");
#endif
#endif
}

// ---------------------------------------------------------------------------
// fp32 WMMA: D = A(16x4) * B(4x16) + C(16x16), wave32
// ---------------------------------------------------------------------------
__device__ __forceinline__ v8f wmma4(v2f a, v2f b, v8f c) {
  return __builtin_amdgcn_wmma_f32_16x16x4_f32(
      /*neg_a=*/false, a, /*neg_b=*/false, b,
      /*c_mod=*/(short)0, c, /*reuse_a=*/false, /*reuse_b=*/false);
}

// A-frag (16x4) from row-major [M x K] storage.
// ISA layout: lanes 0-15 hold K=k0,k0+1 ; lanes 16-31 hold K=k0+2,k0+3 ; M = lane&15.
__device__ __forceinline__ v2f frag_a(const float* base, int stride, int m0, int k0) {
  int lane = threadIdx.x & 31;
  const float* p = base + (m0 + (lane & 15)) * stride + k0 + ((lane >> 4) << 1);
  v2f r; r.x = p[0]; r.y = p[1];
  return r;
}

// B-frag (4x16) where B[k][n] is stored N-major (i.e. Bt[n][k], contiguous in k).
__device__ __forceinline__ v2f frag_bt(const float* base, int stride, int n0, int k0) {
  int lane = threadIdx.x & 31;
  const float* p = base + (n0 + (lane & 15)) * stride + k0 + ((lane >> 4) << 1);
  v2f r; r.x = p[0]; r.y = p[1];
  return r;
}

// B-frag (4x16) where B is stored row-major [K x N].
__device__ __forceinline__ v2f frag_b(const float* base, int stride, int k0, int n0) {
  int lane = threadIdx.x & 31;
  int kk = k0 + ((lane >> 4) << 1);
  int n  = n0 + (lane & 15);
  v2f r; r.x = base[kk * stride + n]; r.y = base[(kk + 1) * stride + n];
  return r;
}

// C/D store: VGPR g -> row m0+g (lanes 0-15) / m0+g+8 (lanes 16-31), col n0+(lane&15)
__device__ __forceinline__ void store_cd(float* base, int stride, int m0, int n0, v8f c) {
  int lane = threadIdx.x & 31;
  int half = lane >> 4, nc = lane & 15;
#pragma unroll
  for (int g = 0; g < 8; ++g)
    base[(m0 + g + 8 * half) * stride + n0 + nc] = c[g];
}

// ---------------------------------------------------------------------------
// Kernel 0: zero the padding row (t=127) of the K/V bh-major buffers
// ---------------------------------------------------------------------------
__global__ void zero_pad_kernel(float* k_ws, float* v_ws) {
  int i = blockIdx.x * 256 + threadIdx.x;   // 64 bh * 64 d
  if (i < NBH * ND) {
    int bh = i >> 6, dd = i & 63;
    k_ws[(bh * 128 + 127) * ND + dd] = 0.f;
    v_ws[(bh * 128 + 127) * ND + dd] = 0.f;
  }
}

// ---------------------------------------------------------------------------
// Kernel 1/4: generic 512-K GEMM  Y = act(X @ W^T + bias)
// 64x64 tile per 256-thread block; modes select A-gather and Y-scatter.
// ---------------------------------------------------------------------------
#define MODE_Q 0
#define MODE_K 1
#define MODE_V 2
#define MODE_O 3

#define XS_STR 36  // pad: conflict-free for 16-row frag reads, 16B-aligned rows

__global__ void gemm512_kernel(const float* __restrict__ A,
                               const float* __restrict__ W,
                               const float* __restrict__ bias,
                               float* __restrict__ Y,
                               int M, float scale, int mode) {
  __shared__ float Xs[64 * XS_STR];
  __shared__ float Ws[64 * XS_STR];
  int tid = threadIdx.x;
  int tm = blockIdx.x, tn = blockIdx.y;
  int w = tid >> 5;
  int tr = w >> 1;            // 0..3: 16-row subtile
  int tc = (w & 1) * 2;       // 0 or 2: two 16-col subtiles per wave
  v8f acc0 = {}, acc1 = {};

  for (int kc = 0; kc < NE; kc += 32) {
    // cooperative staging: 64 rows x 8 float4 each for X and W (async -> LDS)
    for (int i = tid; i < 64 * 8; i += 256) {
      int r = i >> 3, c4 = (i & 7) * 4;
      int m = tm * 64 + r;
      int kk = kc + c4;
      if (mode == MODE_O) {
        // A[m][kk] = ctx_ws[(b*8+h)][t][dd],  m = t*8+b, kk = h*64+dd
        int t = m >> 3, b = m & 7;
        int h = kk >> 6, dd = kk & 63;
        g2l_b128(A + (((b << 3) + h) * 64 + t) * 64 + dd, Xs + r * XS_STR + c4);
      } else if (m < M) {
        g2l_b128(A + m * NE + kk, Xs + r * XS_STR + c4);
      } else {
        *(float4*)(Xs + r * XS_STR + c4) = make_float4(0.f, 0.f, 0.f, 0.f);
      }
      g2l_b128(W + (tn * 64 + r) * NE + kk, Ws + r * XS_STR + c4);
    }
    async_wait0();
    __syncthreads();
#pragma unroll
    for (int kk = 0; kk < 32; kk += 4) {
      v2f a  = frag_a(Xs, XS_STR, tr * 16, kk);
      v2f b0 = frag_bt(Ws, XS_STR, tc * 16, kk);        // B[k][n] = W[n][k]
      v2f b1 = frag_bt(Ws, XS_STR, (tc + 1) * 16, kk);
      acc0 = wmma4(a, b0, acc0);
      acc1 = wmma4(a, b1, acc1);
    }
    __syncthreads();
  }

  // epilogue: bias (+scale) and mode-dependent scatter
  int lane = tid & 31;
  int half = lane >> 4, nc = lane & 15;
#pragma unroll
  for (int g = 0; g < 8; ++g) {
    int m = tm * 64 + tr * 16 + g + 8 * half;
#pragma unroll
    for (int s = 0; s < 2; ++s) {
      int n = tn * 64 + (tc + s) * 16 + nc;
      float v = (s == 0 ? acc0[g] : acc1[g]) + bias[n];
      if (mode == MODE_Q) {
        v *= scale;
        int t = m >> 3, b = m & 7, h = n >> 6, dd = n & 63;
        Y[(((b << 3) + h) * 64 + t) * 64 + dd] = v;
      } else if (mode == MODE_K || mode == MODE_V) {
        if (m < M) {
          int t = m >> 3, b = m & 7, h = n >> 6, dd = n & 63;
          Y[(((b << 3) + h) * 128 + t) * 64 + dd] = v;
        }
      } else {
        Y[m * NE + n] = v;  // out[t][b][e]
      }
    }
  }
}

// ---------------------------------------------------------------------------
// Kernel 2: per-bh scores = q[64x64] @ k[128x64]^T  -> scores_ws[bh][64][128]
// ---------------------------------------------------------------------------
#define SK_STR 68  // 16B-aligned rows; bank rotation 4*r -> conflict-free

__global__ void scores_kernel(const float* __restrict__ q_ws,
                              const float* __restrict__ k_ws,
                              float* __restrict__ scores_ws) {
  int bh = blockIdx.x;
  __shared__ float qs[64 * SK_STR];
  __shared__ float ks[128 * SK_STR];
  int tid = threadIdx.x;
  for (int i = tid; i < 64 * 16; i += 256) {           // float4 units
    int r = i >> 4, c4 = (i & 15) * 4;
    g2l_b128(q_ws + bh * 4096 + r * 64 + c4, qs + r * SK_STR + c4);
  }
  for (int i = tid; i < 128 * 16; i += 256) {
    int r = i >> 4, c4 = (i & 15) * 4;
    g2l_b128(k_ws + bh * 8192 + r * 64 + c4, ks + r * SK_STR + c4);
  }
  async_wait0();
  __syncthreads();

  int w = tid >> 5;
  int tr = w >> 1;            // 4 row tiles
  int tcb = (w & 1) * 4;      // 8 col tiles, 4 per wave
  v8f acc[4] = {};
  for (int k0 = 0; k0 < 64; k0 += 4) {
    v2f a = frag_a(qs, SK_STR, tr * 16, k0);
#pragma unroll
    for (int t = 0; t < 4; ++t) {
      v2f bf = frag_bt(ks, SK_STR, (tcb + t) * 16, k0);  // B[k][j] = k[j][k]
      acc[t] = wmma4(a, bf, acc[t]);
    }
  }
  float* out = scores_ws + bh * (64 * 128);
#pragma unroll
  for (int t = 0; t < 4; ++t)
    store_cd(out, 128, tr * 16, (tcb + t) * 16, acc[t]);
}

// ---------------------------------------------------------------------------
// Kernel 3: per (bh,q): scatter scores into DP table, cumsums,
//           dp = acc_fw @ acc_bw^T (64x64x64 WMMA), gather + softmax -> attn
// ---------------------------------------------------------------------------
__global__ void dp_attn_kernel(const float* __restrict__ scores_ws,
                               const int* __restrict__ indices,
                               float* __restrict__ attn_ws) {
  int bh = blockIdx.x;
  int q  = blockIdx.y;
  int b  = bh >> 3;
  __shared__ float mat[64 * 66];   // scatter target, later dp output
  __shared__ float colc[64 * 66];  // column cumsum (== acc_bw^T as Bt)
  __shared__ float afw[64 * 66];   // row cumsum (acc_fw)
  __shared__ float sc[128];
  __shared__ int rid[128], cid[128];
  __shared__ float s_mx, s_sum;
  int tid = threadIdx.x;

  if (tid < TKN) {
    rid[tid] = indices[(b * TKN + tid) * 2 + 0];
    cid[tid] = indices[(b * TKN + tid) * 2 + 1];
    sc[tid]  = scores_ws[(bh * 64 + q) * 128 + tid];
  }
  for (int i = tid; i < 64 * 66; i += 256) mat[i] = 0.f;
  __syncthreads();
  if (tid < TKN) mat[rid[tid] * 66 + cid[tid]] = sc[tid];
  __syncthreads();

  // column cumsum (threads 0..63) and row cumsum (threads 64..127), both read-only on mat
  if (tid < 64) {
    float s = 0.f;
    for (int r = 0; r < 64; ++r) { s += mat[r * 66 + tid]; colc[r * 66 + tid] = s; }
  } else if (tid < 128) {
    int r = tid - 64;
    float s = 0.f;
    for (int c = 0; c < 64; ++c) { s += mat[r * 66 + c]; afw[r * 66 + c] = s; }
  }
  __syncthreads();

  // dp[r][c] = sum_m afw[r][m] * colc[c][m]  (B[k=m][n=c] read N-major from colc)
  int w = tid >> 5;
  int tr = w >> 1;
  int tc = (w & 1) * 2;
  v8f acc0 = {}, acc1 = {};
  for (int k0 = 0; k0 < 64; k0 += 4) {
    v2f a  = frag_a(afw, 66, tr * 16, k0);
    v2f b0 = frag_bt(colc, 66, tc * 16, k0);
    v2f b1 = frag_bt(colc, 66, (tc + 1) * 16, k0);
    acc0 = wmma4(a, b0, acc0);
    acc1 = wmma4(a, b1, acc1);
  }
  __syncthreads();           // mat reads done; reuse as dp output
  store_cd(mat, 66, tr * 16, tc * 16, acc0);
  store_cd(mat, 66, tr * 16, (tc + 1) * 16, acc1);
  __syncthreads();

  // gather 127 dp entries + softmax
  if (tid < TKN) sc[tid] = mat[rid[tid] * 66 + cid[tid]];
  if (tid == TKN) sc[tid] = -3.0e38f;
  __syncthreads();
  if (tid == 0) {
    float m = -3.0e38f;
    for (int j = 0; j < TKN; ++j) m = fmaxf(m, sc[j]);
    s_mx = m;
  }
  __syncthreads();
  float ex = (tid < TKN) ? __expf(sc[tid] - s_mx) : 0.f;
  if (tid < 128) sc[tid] = ex;   // sc[127] = 0 pads K to 128
  __syncthreads();
  if (tid == 0) {
    float s = 0.f;
    for (int j = 0; j < 128; ++j) s += sc[j];
    s_sum = s;
  }
  __syncthreads();
  if (tid < 128) attn_ws[(bh * 64 + q) * 128 + tid] = sc[tid] / s_sum;
}

// ---------------------------------------------------------------------------
// Kernel 4: per-bh ctx = attn[64x128] @ v[128x64]
// ---------------------------------------------------------------------------
__global__ void ctx_kernel(const float* __restrict__ attn_ws,
                           const float* __restrict__ v_ws,
                           float* __restrict__ ctx_ws) {
  int bh = blockIdx.x;
  __shared__ float vs[128 * SK_STR];
  int tid = threadIdx.x;
  for (int i = tid; i < 128 * 16; i += 256) {          // float4 units
    int r = i >> 4, c4 = (i & 15) * 4;
    g2l_b128(v_ws + bh * 8192 + r * 64 + c4, vs + r * SK_STR + c4);
  }
  async_wait0();
  __syncthreads();

  int w = tid >> 5;
  int tr = w >> 1;
  int tc = (w & 1) * 2;
  v8f acc0 = {}, acc1 = {};
  const float* abase = attn_ws + bh * (64 * 128);  // A-frags straight from L2
  for (int k0 = 0; k0 < 128; k0 += 4) {
    v2f a  = frag_a(abase, 128, tr * 16, k0);
    v2f b0 = frag_b(vs, SK_STR, k0, tc * 16);
    v2f b1 = frag_b(vs, SK_STR, k0, (tc + 1) * 16);
    acc0 = wmma4(a, b0, acc0);
    acc1 = wmma4(a, b1, acc1);
  }
  float* out = ctx_ws + bh * 4096;
  store_cd(out, 64, tr * 16, tc * 16, acc0);
  store_cd(out, 64, tr * 16, (tc + 1) * 16, acc1);
}

// ---------------------------------------------------------------------------
extern "C" void kernel_launch(void* const* d_in, const int* in_sizes, int n_in,
                              void* d_out, int out_size, void* d_ws, size_t ws_size,
                              hipStream_t stream) {
  const float* query = (const float*)d_in[0];   // [64,8,512]
  const float* key   = (const float*)d_in[1];   // [127,8,512]
  const int*   idx   = (const int*)d_in[2];     // [8,127,2]
  const float* Wi    = (const float*)d_in[3];   // [1536,512]
  const float* bi    = (const float*)d_in[4];   // [1536]
  const float* Wo    = (const float*)d_in[5];   // [512,512]
  const float* bo    = (const float*)d_in[6];   // [512]
  float* out = (float*)d_out;                   // [64,8,512]

  float* ws    = (float*)d_ws;
  float* q_ws  = ws;                  // [64][64][64]   = 262144
  float* k_ws  = q_ws + 262144;       // [64][128][64]  = 524288
  float* v_ws  = k_ws + 524288;       // [64][128][64]  = 524288
  float* sc_ws = v_ws + 524288;       // [64][64][128]  = 524288
  float* at_ws = sc_ws + 524288;      // [64][64][128]  = 524288
  float* ct_ws = at_ws + 524288;      // [64][64][64]   = 262144

  const float scaling = 0.125f;       // d^-0.5 = 64^-0.5

  zero_pad_kernel<<<16, 256, 0, stream>>>(k_ws, v_ws);

  gemm512_kernel<<<dim3(8, 8),  256, 0, stream>>>(query, Wi,              bi,        q_ws, 512,  scaling, MODE_Q);
  gemm512_kernel<<<dim3(16, 8), 256, 0, stream>>>(key,   Wi + 512 * 512,  bi + 512,  k_ws, 1016, 1.f,     MODE_K);
  gemm512_kernel<<<dim3(16, 8), 256, 0, stream>>>(key,   Wi + 1024 * 512, bi + 1024, v_ws, 1016, 1.f,     MODE_V);

  scores_kernel<<<64, 256, 0, stream>>>(q_ws, k_ws, sc_ws);
  dp_attn_kernel<<<dim3(64, 64), 256, 0, stream>>>(sc_ws, idx, at_ws);
  ctx_kernel<<<64, 256, 0, stream>>>(at_ws, v_ws, ct_ws);

  gemm512_kernel<<<dim3(8, 8), 256, 0, stream>>>(ct_ws, Wo, bo, out, 512, 1.f, MODE_O);
}